// ExpCloudMMD_53025666236933
// MI455X (gfx1250) — compile-verified
//
#include <hip/hip_runtime.h>

typedef __attribute__((ext_vector_type(2))) float v2f;
typedef __attribute__((ext_vector_type(8))) float v8f;

#define GAMMA 0.5f

// ---------------------------------------------------------------------------
// Row squared-norms for [N, 16] f32 matrix: out[r] = sum_k A[r][k]^2
// ---------------------------------------------------------------------------
__global__ void rownorm16(const float* __restrict__ A, float* __restrict__ out, int nrows)
{
    int r = blockIdx.x * blockDim.x + threadIdx.x;
    if (r < nrows) {
        const float4* p = (const float4*)(A + (size_t)r * 16);
        float s = 0.0f;
        #pragma unroll
        for (int i = 0; i < 4; ++i) {
            float4 v = p[i];
            s += v.x * v.x + v.y * v.y + v.z * v.z + v.w * v.w;
        }
        out[r] = s;
    }
}

// ---------------------------------------------------------------------------
// One wave (32 lanes) computes a 16-row M-tile of the RBF Gram matrix against
// a chunk of 16-col N-tiles, using V_WMMA_F32_16X16X4_F32 chained 4x (K=16).
//
// Fragment layout (CDNA5 ISA, 32-bit A 16x4): for sub-block kb (K = 4*kb..4*kb+3),
// lane = (h<<4)|lr holds row (m0+lr), elements K = 4*kb + 2*h + {0,1}  -> one
// aligned float2 load per fragment. B (4x16 of P^T) has the identical formula
// on P's rows since B[k][n] = P[n][k].
//
// C/D layout: VGPR r at lane (h,lr) holds element (M = 8*h + r, N = lr).
//
// partials[block] = coeff * sum_tile exp(-GAMMA * max(a2 + b2 - 2*dot, 0))
// ---------------------------------------------------------------------------
__global__ __launch_bounds__(32)
void gram_exp_sum(const float* __restrict__ A, const float* __restrict__ B,
                  const float* __restrict__ a2, const float* __restrict__ b2,
                  int Ntiles, float coeff, float* __restrict__ partials)
{
    const int lane = threadIdx.x;     // 0..31, full wave active (WMMA needs EXEC all 1s)
    const int lr   = lane & 15;
    const int h    = lane >> 4;
    const int m0   = blockIdx.x * 16;

    const int tiles_per_chunk = Ntiles / gridDim.y;
    const int nt0 = blockIdx.y * tiles_per_chunk;
    const int nt1 = nt0 + tiles_per_chunk;

    // A fragments: loaded once, reused for all N-tiles in the chunk.
    const float* Ap = A + (size_t)(m0 + lr) * 16 + 2 * h;
    const v2f a0 = *(const v2f*)(Ap + 0);
    const v2f a1 = *(const v2f*)(Ap + 4);
    const v2f a2f = *(const v2f*)(Ap + 8);
    const v2f a3 = *(const v2f*)(Ap + 12);

    // Per-lane row norms for the 8 C rows this lane holds (M = 8*h + r).
    float a2row[8];
    #pragma unroll
    for (int r = 0; r < 8; ++r) a2row[r] = a2[m0 + 8 * h + r];

    float acc = 0.0f;

    for (int nt = nt0; nt < nt1; ++nt) {
        const int n0 = nt * 16;
        const float* Bp = B + (size_t)(n0 + lr) * 16 + 2 * h;
        const v2f b0 = *(const v2f*)(Bp + 0);
        const v2f b1 = *(const v2f*)(Bp + 4);
        const v2f b2v = *(const v2f*)(Bp + 8);
        const v2f b3 = *(const v2f*)(Bp + 12);

        v8f c = {};
        // D = A x B + C, 8 args: (neg_a, A, neg_b, B, c_mod, C, reuse_a, reuse_b)
        c = __builtin_amdgcn_wmma_f32_16x16x4_f32(false, a0,  false, b0,  (short)0, c, false, false);
        c = __builtin_amdgcn_wmma_f32_16x16x4_f32(false, a1,  false, b1,  (short)0, c, false, false);
        c = __builtin_amdgcn_wmma_f32_16x16x4_f32(false, a2f, false, b2v, (short)0, c, false, false);
        c = __builtin_amdgcn_wmma_f32_16x16x4_f32(false, a3,  false, b3,  (short)0, c, false, false);

        const float b2l = b2[n0 + lr];
        #pragma unroll
        for (int r = 0; r < 8; ++r) {
            float sq = fmaxf(a2row[r] + b2l - 2.0f * c[r], 0.0f);
            acc += __expf(-GAMMA * sq);
        }
    }

    // wave32 butterfly reduction
    #pragma unroll
    for (int off = 16; off > 0; off >>= 1)
        acc += __shfl_xor(acc, off, 32);

    if (lane == 0)
        partials[(size_t)blockIdx.x * gridDim.y + blockIdx.y] = coeff * acc;
}

// ---------------------------------------------------------------------------
// Deterministic final reduction of all partial sums into out[0].
// ---------------------------------------------------------------------------
__global__ void reduce_sum(const float* __restrict__ partials, int n, float* __restrict__ out)
{
    __shared__ float sm[256];
    float s = 0.0f;
    for (int i = threadIdx.x; i < n; i += 256) s += partials[i];
    sm[threadIdx.x] = s;
    __syncthreads();
    #pragma unroll
    for (int w = 128; w > 0; w >>= 1) {
        if ((int)threadIdx.x < w) sm[threadIdx.x] += sm[threadIdx.x + w];
        __syncthreads();
    }
    if (threadIdx.x == 0) out[0] = sm[0];
}

// ---------------------------------------------------------------------------
// result = (1/Np^2) * sum K(P,P)  -  (2/(Nx*Np)) * sum K(X,P)
// ---------------------------------------------------------------------------
extern "C" void kernel_launch(void* const* d_in, const int* in_sizes, int n_in,
                              void* d_out, int out_size, void* d_ws, size_t ws_size,
                              hipStream_t stream)
{
    const float* x = (const float*)d_in[0];     // [Nx, 16]
    const float* p = (const float*)d_in[1];     // [Np, 16]
    float* out = (float*)d_out;

    const int D  = 16;
    const int Nx = in_sizes[0] / D;             // 32768
    const int Np = in_sizes[1] / D;             // 8192

    // workspace: [x2 | p2 | partials]
    float* x2       = (float*)d_ws;
    float* p2       = x2 + Nx;
    float* partials = p2 + Np;

    rownorm16<<<dim3((Nx + 255) / 256), dim3(256), 0, stream>>>(x, x2, Nx);
    rownorm16<<<dim3((Np + 255) / 256), dim3(256), 0, stream>>>(p, p2, Np);

    const int NCH = 16;                         // N-chunks per M-tile (parallelism knob)
    const int MtP = Np / 16;                    // 512  M-tiles (PP)
    const int MtX = Nx / 16;                    // 2048 M-tiles (XP)
    const int Nt  = Np / 16;                    // 512  N-tiles

    const int nPartPP = MtP * NCH;              // 8192
    const int nPartXP = MtX * NCH;              // 32768

    const float c1 =  1.0f / ((float)Np * (float)Np);
    const float c2 = -2.0f / ((float)Nx * (float)Np);

    gram_exp_sum<<<dim3(MtP, NCH), dim3(32), 0, stream>>>(p, p, p2, p2, Nt, c1, partials);
    gram_exp_sum<<<dim3(MtX, NCH), dim3(32), 0, stream>>>(x, p, x2, p2, Nt, c2, partials + nPartPP);

    reduce_sum<<<dim3(1), dim3(256), 0, stream>>>(partials, nPartPP + nPartXP, out);
}